// GCNN_58050777973328
// MI455X (gfx1250) — compile-verified
//
#include <hip/hip_runtime.h>

#define NNODES 100000
#define NEDGES 3200000
#define NB 64
#define NFEAT 21
#define SLOPE 0.01f

typedef __attribute__((ext_vector_type(2))) float v2f;
typedef __attribute__((ext_vector_type(8))) float v8f;

__global__ void k_fill(float* __restrict__ p, float v, int n) {
    int i = blockIdx.x * blockDim.x + threadIdx.x;
    if (i < n) p[i] = v;
}

// deg[dst[e]] += ew[e]   (deg pre-filled with 1.0 = self-loop weight)
__global__ void k_deg(const int* __restrict__ dst, const float* __restrict__ ew,
                      float* __restrict__ deg, int e_cnt) {
    int e = blockIdx.x * blockDim.x + threadIdx.x;
    if (e < e_cnt)
        __hip_atomic_fetch_add(&deg[dst[e]], ew[e], __ATOMIC_RELAXED, __HIP_MEMORY_SCOPE_AGENT);
}

// in-place deg -> dinv
__global__ void k_dinv(float* __restrict__ d, int n) {
    int i = blockIdx.x * blockDim.x + threadIdx.x;
    if (i < n) { float v = d[i]; d[i] = (v > 0.f) ? rsqrtf(v) : 0.f; }
}

// C[M x Nn] = act(A[M x K] @ W[K x Nn] + bias)
// One wave per 16x16 tile via V_WMMA_F32_16X16X4_F32 (fp32, matches reference).
// Requires M % 16 == 0 (true for all call sites). act: 0=none, 1=leaky relu.
// The 16-wide W slice for this n-tile is staged in LDS, zero-padded to
// Kpad = ceil(K/4)*4 rows, so the inner loop has NO bounds checks on B and
// the main K loop has NO bounds checks at all (tail chunk guards A only).
__global__ void k_gemm_wmma(const float* __restrict__ A, const float* __restrict__ W,
                            const float* __restrict__ bias, float* __restrict__ C,
                            int K, int Nn, int act) {
    extern __shared__ float ldsB[];              // [Kpad][16]
    const int lane = threadIdx.x & 31;
    const int hk   = lane >> 4;                  // 0: K pair {0,1}; 1: K pair {2,3}
    const int l    = lane & 15;
    const int m0 = blockIdx.x * 16;
    const int n0 = blockIdx.y * 16;
    const int n  = n0 + l;
    const int Kpad = (K + 3) & ~3;

    // stage zero-padded W[:, n0:n0+16] into LDS (coalesced 16-wide rows)
    for (int i = lane; i < Kpad * 16; i += 32) {
        int k  = i >> 4;
        int nn = n0 + (i & 15);
        ldsB[i] = (k < K && nn < Nn) ? W[(size_t)k * Nn + nn] : 0.f;
    }
    __syncthreads();

    const float* arow = A + (size_t)(m0 + l) * K;   // A lane row (M = l for both halves)
    v8f acc = {};
    int k0 = 0;
    for (; k0 + 4 <= K; k0 += 4) {                  // guard-free main loop
        int ka = k0 + hk * 2;
        v2f av, bv;
        av.x = arow[ka];
        av.y = arow[ka + 1];
        bv.x = ldsB[ka * 16 + l];
        bv.y = ldsB[(ka + 1) * 16 + l];
        acc = __builtin_amdgcn_wmma_f32_16x16x4_f32(false, av, false, bv,
                                                    (short)0, acc, false, false);
    }
    if (k0 < K) {                                   // tail chunk (K % 4 != 0)
        int ka = k0 + hk * 2;
        v2f av, bv;
        av.x = (ka     < K) ? arow[ka]     : 0.f;
        av.y = (ka + 1 < K) ? arow[ka + 1] : 0.f;
        bv.x = ldsB[ka * 16 + l];                   // LDS already zero-padded
        bv.y = ldsB[(ka + 1) * 16 + l];
        acc = __builtin_amdgcn_wmma_f32_16x16x4_f32(false, av, false, bv,
                                                    (short)0, acc, false, false);
    }

    if (n < Nn) {
        float bb = bias ? bias[n] : 0.f;
        for (int r = 0; r < 8; ++r) {
            int row = m0 + hk * 8 + r;   // C layout: vgpr r -> M=r (lanes 0-15), M=8+r (16-31)
            float v = acc[r] + bb;
            if (act) v = (v >= 0.f) ? v : SLOPE * v;
            C[(size_t)row * Nn + n] = v;
        }
    }
}

// out[n,f] = dinv[n]^2 * xw[n,f] + b[f]    (self-loop contribution + bias)
__global__ void k_agg_init(const float* __restrict__ xw, const float* __restrict__ dinv,
                           const float* __restrict__ bias, float* __restrict__ out, int total) {
    int i = blockIdx.x * blockDim.x + threadIdx.x;
    if (i < total) {
        int nidx = i / NFEAT;
        int f    = i - nidx * NFEAT;
        float di = dinv[nidx];
        out[i] = di * di * xw[i] + bias[f];
    }
}

// out[dst,f] += dinv[src]*ew*dinv[dst] * xw[src,f]  ; thread = (edge, feature)
__global__ void k_edge(const int* __restrict__ src, const int* __restrict__ dst,
                       const float* __restrict__ ew, const float* __restrict__ dinv,
                       const float* __restrict__ xw, float* __restrict__ out, int total) {
    int i = blockIdx.x * blockDim.x + threadIdx.x;
    if (i < total) {
        int e = i / NFEAT;
        int f = i - e * NFEAT;
        int s = src[e], d = dst[e];
        float coef = dinv[s] * ew[e] * dinv[d];
        __hip_atomic_fetch_add(&out[(size_t)d * NFEAT + f], coef * xw[(size_t)s * NFEAT + f],
                               __ATOMIC_RELAXED, __HIP_MEMORY_SCOPE_AGENT);
    }
}

// hierarchical relu + segment-sum pool: LDS partials, then global flush of nonzeros
__global__ void k_pool(const float* __restrict__ x, const int* __restrict__ batch,
                       float* __restrict__ gsum, float* __restrict__ gcnt, int n_nodes) {
    __shared__ float sp[NB * NFEAT];
    __shared__ float sc[NB];
    for (int i = threadIdx.x; i < NB * NFEAT; i += blockDim.x) sp[i] = 0.f;
    for (int i = threadIdx.x; i < NB;         i += blockDim.x) sc[i] = 0.f;
    __syncthreads();
    int nidx = blockIdx.x * blockDim.x + threadIdx.x;
    if (nidx < n_nodes) {
        int b = batch[nidx];
        for (int f = 0; f < NFEAT; ++f) {
            float v = x[(size_t)nidx * NFEAT + f];
            v = (v > 0.f) ? v : 0.f;                 // relu before pooling
            atomicAdd(&sp[b * NFEAT + f], v);
        }
        atomicAdd(&sc[b], 1.f);
    }
    __syncthreads();
    for (int i = threadIdx.x; i < NB * NFEAT; i += blockDim.x)
        if (sp[i] != 0.f)
            __hip_atomic_fetch_add(&gsum[i], sp[i], __ATOMIC_RELAXED, __HIP_MEMORY_SCOPE_AGENT);
    for (int i = threadIdx.x; i < NB; i += blockDim.x)
        if (sc[i] != 0.f)
            __hip_atomic_fetch_add(&gcnt[i], sc[i], __ATOMIC_RELAXED, __HIP_MEMORY_SCOPE_AGENT);
}

__global__ void k_pool_final(const float* __restrict__ gsum, const float* __restrict__ gcnt,
                             float* __restrict__ pooled) {
    int i = blockIdx.x * blockDim.x + threadIdx.x;
    if (i < NB * NFEAT) {
        int b = i / NFEAT;
        pooled[i] = gsum[i] / fmaxf(gcnt[b], 1.f);
    }
}

__global__ void k_softmax5(const float* __restrict__ logits, float* __restrict__ out) {
    int r = threadIdx.x;
    if (r < NB) {
        float m = logits[r * 5];
        for (int f = 1; f < 5; ++f) m = fmaxf(m, logits[r * 5 + f]);
        float e[5], s = 0.f;
        for (int f = 0; f < 5; ++f) { e[f] = expf(logits[r * 5 + f] - m); s += e[f]; }
        float inv = 1.f / s;
        for (int f = 0; f < 5; ++f) out[r * 5 + f] = e[f] * inv;
    }
}

static inline unsigned gemm_smem_bytes(int K) {
    int Kpad = (K + 3) & ~3;
    return (unsigned)(Kpad * 16 * sizeof(float));
}

extern "C" void kernel_launch(void* const* d_in, const int* in_sizes, int n_in,
                              void* d_out, int out_size, void* d_ws, size_t ws_size,
                              hipStream_t stream) {
    const float* x0    = (const float*)d_in[0];
    const int*   ei    = (const int*)  d_in[1];
    const float* ew    = (const float*)d_in[2];
    const int*   batch = (const int*)  d_in[3];
    const float* W1 = (const float*)d_in[4];  const float* b1 = (const float*)d_in[5];
    const float* W2 = (const float*)d_in[6];  const float* b2 = (const float*)d_in[7];
    const float* W3 = (const float*)d_in[8];  const float* b3 = (const float*)d_in[9];
    const float* Wp  = (const float*)d_in[10]; const float* bp  = (const float*)d_in[11];
    const float* Wf1 = (const float*)d_in[12]; const float* bf1 = (const float*)d_in[13];
    const float* Wf2 = (const float*)d_in[14]; const float* bf2 = (const float*)d_in[15];
    const float* Wo  = (const float*)d_in[16]; const float* bo  = (const float*)d_in[17];

    const int* src = ei;
    const int* dst = ei + NEDGES;

    // workspace carve-up (floats)
    float* ws     = (float*)d_ws;
    float* dinv   = ws;                                    // N
    float* bufA   = dinv + NNODES;                         // N*21  (xw)
    float* bufB   = bufA + (size_t)NNODES * NFEAT;         // N*21  (layer out / next x)
    float* gsum   = bufB + (size_t)NNODES * NFEAT;         // 64*21
    float* gcnt   = gsum + NB * NFEAT;                     // 64
    float* pooled = gcnt + NB;                             // 64*21
    float* h1     = pooled + NB * NFEAT;                   // 64*128
    float* h2     = h1 + NB * 128;                         // 64*256
    float* h3     = h2 + NB * 256;                         // 64*64
    float* logits = h3 + NB * 64;                          // 64*5

    const int TB = 256;
    // normalization (layer-invariant): deg = 1 + scatter(ew by dst) -> rsqrt
    k_fill<<<(NNODES + TB - 1) / TB, TB, 0, stream>>>(dinv, 1.0f, NNODES);
    k_fill<<<(NB * NFEAT + NB + TB - 1) / TB, TB, 0, stream>>>(gsum, 0.f, NB * NFEAT + NB);
    k_deg<<<(NEDGES + TB - 1) / TB, TB, 0, stream>>>(dst, ew, dinv, NEDGES);
    k_dinv<<<(NNODES + TB - 1) / TB, TB, 0, stream>>>(dinv, NNODES);

    const float* Ws[3] = {W1, W2, W3};
    const float* bs[3] = {b1, b2, b3};
    const float* xin = x0;
    const int node_tot = NNODES * NFEAT;       // 2.1M
    const int edge_tot = NEDGES * NFEAT;       // 67.2M
    for (int l = 0; l < 3; ++l) {
        dim3 g(NNODES / 16, 2);                // Nn=21 -> two 16-wide n-tiles
        k_gemm_wmma<<<g, 32, gemm_smem_bytes(NFEAT), stream>>>(xin, Ws[l], nullptr, bufA,
                                                               NFEAT, NFEAT, 0);
        k_agg_init<<<(node_tot + TB - 1) / TB, TB, 0, stream>>>(bufA, dinv, bs[l], bufB, node_tot);
        k_edge<<<(edge_tot + TB - 1) / TB, TB, 0, stream>>>(src, dst, ew, dinv, bufA, bufB, edge_tot);
        xin = bufB;
    }

    k_pool<<<(NNODES + TB - 1) / TB, TB, 0, stream>>>(bufB, batch, gsum, gcnt, NNODES);
    k_pool_final<<<(NB * NFEAT + TB - 1) / TB, TB, 0, stream>>>(gsum, gcnt, pooled);

    // MLP head (all M=64 -> 4 m-tiles), WMMA + fused bias/leaky
    { dim3 g(4, 8);  k_gemm_wmma<<<g, 32, gemm_smem_bytes(21),  stream>>>(pooled, Wp,  bp,  h1, 21,  128, 1); }
    { dim3 g(4, 16); k_gemm_wmma<<<g, 32, gemm_smem_bytes(128), stream>>>(h1,     Wf1, bf1, h2, 128, 256, 1); }
    { dim3 g(4, 4);  k_gemm_wmma<<<g, 32, gemm_smem_bytes(256), stream>>>(h2,     Wf2, bf2, h3, 256, 64,  1); }
    { dim3 g(4, 1);  k_gemm_wmma<<<g, 32, gemm_smem_bytes(64),  stream>>>(h3,     Wo,  bo,  logits, 64, 5, 0); }

    k_softmax5<<<1, 64, 0, stream>>>(logits, (float*)d_out);
}